// LoopyBeliefPropagation_54528904790123
// MI455X (gfx1250) — compile-verified
//
#include <hip/hip_runtime.h>

#define N_NODES 10000
#define N_MSGS  320000
#define N_E2E   2560000
#define NODE_F  128
#define EDGE_F  32
#define OUTF    64
#define STEPS   3

typedef __attribute__((ext_vector_type(2))) float v2f;
typedef __attribute__((ext_vector_type(8))) float v8f;

// ---------------------------------------------------------------------------
// Kernel 1: input_message = node_feat[n2e_src] @ W_n2l + edge_feat @ W_e2l
//           message       = relu(input_message)
// One wave = one 16x16 output tile, fp32 WMMA K=4 steps. 8 waves/block.
// ---------------------------------------------------------------------------
__global__ __launch_bounds__(256) void k_init(const float* __restrict__ node_feat,
                                              const float* __restrict__ edge_feat,
                                              const float* __restrict__ Wn,
                                              const float* __restrict__ We,
                                              const int*   __restrict__ n2e_src,
                                              float* __restrict__ im,
                                              float* __restrict__ msg)
{
    __shared__ float lw[NODE_F * OUTF + EDGE_F * OUTF];   // 40 KB
    const int tid = threadIdx.x;
    for (int i = tid; i < NODE_F * OUTF + EDGE_F * OUTF; i += 256)
        lw[i] = (i < NODE_F * OUTF) ? Wn[i] : We[i - NODE_F * OUTF];
    __syncthreads();

    const int wave = tid >> 5;
    const int lane = tid & 31;
    const int lid  = lane & 15;
    const int h    = lane >> 4;
    const int row_base = (blockIdx.x * 2 + (wave >> 2)) * 16;
    const int n0 = (wave & 3) * 16;
    const int myrow = row_base + lid;

    const float* Arow = node_feat + (size_t)n2e_src[myrow] * NODE_F;
    v8f acc = {0.f, 0.f, 0.f, 0.f, 0.f, 0.f, 0.f, 0.f};

    #pragma unroll
    for (int kk = 0; kk < NODE_F; kk += 4) {
        v2f a = *(const v2f*)(Arow + kk + 2 * h);
        v2f b;
        b.x = lw[(kk + 2 * h)     * OUTF + n0 + lid];
        b.y = lw[(kk + 2 * h + 1) * OUTF + n0 + lid];
        acc = __builtin_amdgcn_wmma_f32_16x16x4_f32(false, a, false, b,
                                                    (short)0, acc, false, false);
    }
    const float* Erow = edge_feat + (size_t)myrow * EDGE_F;
    #pragma unroll
    for (int kk = 0; kk < EDGE_F; kk += 4) {
        v2f a = *(const v2f*)(Erow + kk + 2 * h);
        v2f b;
        b.x = lw[NODE_F * OUTF + (kk + 2 * h)     * OUTF + n0 + lid];
        b.y = lw[NODE_F * OUTF + (kk + 2 * h + 1) * OUTF + n0 + lid];
        acc = __builtin_amdgcn_wmma_f32_16x16x4_f32(false, a, false, b,
                                                    (short)0, acc, false, false);
    }
    #pragma unroll
    for (int r = 0; r < 8; ++r) {
        const int row = row_base + r + 8 * h;
        const size_t idx = (size_t)row * OUTF + n0 + lid;
        const float v = acc[r];
        im[idx]  = v;
        msg[idx] = v > 0.f ? v : 0.f;
    }
}

// ---------------------------------------------------------------------------
// Kernel 2: agg[e2e_dst[e]] += message[e2e_src[e]]   (64-float rows)
// 16 lanes per edge, float4 gather + 4x global_atomic_add_f32 scatter.
// ---------------------------------------------------------------------------
__global__ __launch_bounds__(256) void k_scatter(const float* __restrict__ msg,
                                                 const int*   __restrict__ esrc,
                                                 const int*   __restrict__ edst,
                                                 float* __restrict__ agg)
{
    const size_t t = (size_t)blockIdx.x * 256 + threadIdx.x;
    const int e = (int)(t >> 4);
    const int q = (int)(t & 15);
    const int s = esrc[e];
    const int d = edst[e];
    const float4 v = ((const float4*)msg)[(size_t)s * 16 + q];
    float* p = agg + (size_t)d * 64 + q * 4;
    unsafeAtomicAdd(p + 0, v.x);
    unsafeAtomicAdd(p + 1, v.y);
    unsafeAtomicAdd(p + 2, v.z);
    unsafeAtomicAdd(p + 3, v.w);
}

// ---------------------------------------------------------------------------
// Kernel 3: message = relu(input_message + agg @ W_rec)
// input_message tile preloaded into the WMMA C accumulator.
// ---------------------------------------------------------------------------
__global__ __launch_bounds__(256) void k_recur(const float* __restrict__ agg,
                                               const float* __restrict__ Wr,
                                               const float* __restrict__ im,
                                               float* __restrict__ msg)
{
    __shared__ float lw[OUTF * OUTF];   // 16 KB
    const int tid = threadIdx.x;
    for (int i = tid; i < OUTF * OUTF; i += 256) lw[i] = Wr[i];
    __syncthreads();

    const int wave = tid >> 5;
    const int lane = tid & 31;
    const int lid  = lane & 15;
    const int h    = lane >> 4;
    const int row_base = (blockIdx.x * 2 + (wave >> 2)) * 16;
    const int n0 = (wave & 3) * 16;
    const int myrow = row_base + lid;

    v8f acc;
    #pragma unroll
    for (int r = 0; r < 8; ++r)
        acc[r] = im[(size_t)(row_base + r + 8 * h) * OUTF + n0 + lid];

    const float* Arow = agg + (size_t)myrow * OUTF;
    #pragma unroll
    for (int kk = 0; kk < OUTF; kk += 4) {
        v2f a = *(const v2f*)(Arow + kk + 2 * h);
        v2f b;
        b.x = lw[(kk + 2 * h)     * OUTF + n0 + lid];
        b.y = lw[(kk + 2 * h + 1) * OUTF + n0 + lid];
        acc = __builtin_amdgcn_wmma_f32_16x16x4_f32(false, a, false, b,
                                                    (short)0, acc, false, false);
    }
    #pragma unroll
    for (int r = 0; r < 8; ++r) {
        const float v = acc[r];
        msg[(size_t)(row_base + r + 8 * h) * OUTF + n0 + lid] = v > 0.f ? v : 0.f;
    }
}

// ---------------------------------------------------------------------------
// Kernel 4: out[red_dst[m]] += message[m]
// ---------------------------------------------------------------------------
__global__ __launch_bounds__(256) void k_final(const float* __restrict__ msg,
                                               const int*   __restrict__ rdst,
                                               float* __restrict__ out)
{
    const size_t t = (size_t)blockIdx.x * 256 + threadIdx.x;
    const int m = (int)(t >> 4);
    const int q = (int)(t & 15);
    const int d = rdst[m];
    const float4 v = ((const float4*)msg)[(size_t)m * 16 + q];
    float* p = out + (size_t)d * 64 + q * 4;
    unsafeAtomicAdd(p + 0, v.x);
    unsafeAtomicAdd(p + 1, v.y);
    unsafeAtomicAdd(p + 2, v.z);
    unsafeAtomicAdd(p + 3, v.w);
}

extern "C" void kernel_launch(void* const* d_in, const int* in_sizes, int n_in,
                              void* d_out, int out_size, void* d_ws, size_t ws_size,
                              hipStream_t stream) {
    const float* node_feat = (const float*)d_in[0];
    const float* edge_feat = (const float*)d_in[1];
    const float* Wn        = (const float*)d_in[2];
    const float* We        = (const float*)d_in[3];
    const float* Wr        = (const float*)d_in[4];
    const int*   n2e_src   = (const int*)d_in[5];
    const int*   esrc      = (const int*)d_in[6];
    const int*   edst      = (const int*)d_in[7];
    const int*   rdst      = (const int*)d_in[8];

    float* im  = (float*)d_ws;                    // [N_MSGS, 64]
    float* msg = im  + (size_t)N_MSGS * OUTF;     // [N_MSGS, 64]
    float* agg = msg + (size_t)N_MSGS * OUTF;     // [N_MSGS, 64]

    k_init<<<N_MSGS / 32, 256, 0, stream>>>(node_feat, edge_feat, Wn, We,
                                            n2e_src, im, msg);
    for (int s = 0; s < STEPS; ++s) {
        hipMemsetAsync(agg, 0, (size_t)N_MSGS * OUTF * sizeof(float), stream);
        k_scatter<<<(int)((size_t)N_E2E * 16 / 256), 256, 0, stream>>>(msg, esrc, edst, agg);
        k_recur<<<N_MSGS / 32, 256, 0, stream>>>(agg, Wr, im, msg);
    }
    hipMemsetAsync(d_out, 0, (size_t)N_NODES * OUTF * sizeof(float), stream);
    k_final<<<(int)((size_t)N_MSGS * 16 / 256), 256, 0, stream>>>(msg, rdst, (float*)d_out);
}